// MandiFlowNet_38414187495629
// MI455X (gfx1250) — compile-verified
//
#include <hip/hip_runtime.h>

typedef float v2f __attribute__((ext_vector_type(2)));
typedef float v8f __attribute__((ext_vector_type(8)));

#define D_FEAT 128

// ---------------------------------------------------------------------------
// Tensor Data Mover: async 2D tile load Global -> LDS (6-arg clang builtin).
// Descriptor layout per CDNA5 ISA 8.3/8.4 (D# group0 128b, group1 256b).
// ---------------------------------------------------------------------------
__device__ __forceinline__ void tdm_load_2d(
    unsigned lds_addr, const void* gptr,
    unsigned td0,      // tensor extent dim0 (elements, contiguous dim)
    unsigned td1,      // tensor extent dim1 (rows remaining from tile start)
    unsigned stride0,  // row stride in elements
    unsigned tile0, unsigned tile1,
    unsigned padIntervalCode, unsigned padAmountCode, bool padEn)
{
    typedef unsigned int u32x4 __attribute__((ext_vector_type(4)));
    typedef int i32x8 __attribute__((ext_vector_type(8)));
    typedef int i32x4 __attribute__((ext_vector_type(4)));

    unsigned long long ga = (unsigned long long)(uintptr_t)gptr;
    u32x4 g0;
    g0[0] = 1u;                                           // count=1: valid user D#
    g0[1] = lds_addr;                                     // LDS byte address
    g0[2] = (unsigned)ga;                                 // global_addr[31:0]
    g0[3] = (unsigned)((ga >> 32) & 0x01FFFFFFu)          // global_addr[56:32]
          | (2u << 30);                                   // type = 2 ("image")

    unsigned dw0 = (2u << 16)                             // data_size = 4 bytes
                 | (padEn ? (1u << 20) : 0u)              // pad_enable
                 | ((padIntervalCode & 7u) << 22)         // pad interval (2^(c+1) dwords)
                 | ((padAmountCode & 127u) << 25);        // pad amount (c+1 dwords)

    unsigned long long q0 = (unsigned long long)dw0
                          | ((unsigned long long)(td0 & 0xFFFFu) << 48);       // td0[15:0] @48
    unsigned long long q1 = (unsigned long long)(td0 >> 16)                    // td0[31:16] @64
                          | ((unsigned long long)td1 << 16)                    // td1 @80
                          | ((unsigned long long)(tile0 & 0xFFFFu) << 48);     // tile0 @112
    unsigned long long q2 = (unsigned long long)(tile1 & 0xFFFFu)              // tile1 @128
                          | ((unsigned long long)stride0 << 32);               // stride0[31:0] @160
    unsigned long long q3 = 0ull;                                              // stride0[47:32]=0, dim1_stride unused

    i32x8 g1;
    g1[0] = (int)(unsigned)q0;  g1[1] = (int)(unsigned)(q0 >> 32);
    g1[2] = (int)(unsigned)q1;  g1[3] = (int)(unsigned)(q1 >> 32);
    g1[4] = (int)(unsigned)q2;  g1[5] = (int)(unsigned)(q2 >> 32);
    g1[6] = (int)(unsigned)q3;  g1[7] = (int)(unsigned)(q3 >> 32);

    i32x4 z4 = {0, 0, 0, 0};
    i32x8 z8 = {0, 0, 0, 0, 0, 0, 0, 0};
    __builtin_amdgcn_tensor_load_to_lds(g0, g1, z4, z4, z8, 0);
}

// ---------------------------------------------------------------------------
// Degree helpers: deg[i] = 1 + in_degree(i), then dinv = rsqrt(deg) in place.
// ---------------------------------------------------------------------------
__global__ void init_ones_kernel(float* __restrict__ d, int n) {
    int i = blockIdx.x * blockDim.x + threadIdx.x;
    if (i < n) d[i] = 1.0f;
}

__global__ void count_deg_kernel(const int* __restrict__ dst, float* __restrict__ deg, int e) {
    int i = blockIdx.x * blockDim.x + threadIdx.x;
    if (i < e) unsafeAtomicAdd(&deg[dst[i]], 1.0f);
}

__global__ void rsqrt_kernel(float* __restrict__ d, int n) {
    int i = blockIdx.x * blockDim.x + threadIdx.x;
    if (i < n) d[i] = rsqrtf(d[i]);
}

__global__ void zero4_kernel(float4* __restrict__ p, int n4) {
    int i = blockIdx.x * blockDim.x + threadIdx.x;
    if (i < n4) p[i] = make_float4(0.f, 0.f, 0.f, 0.f);
}

// ---------------------------------------------------------------------------
// GEMM: H[n x 128] = X[n x 128] @ W[128 x 128] using V_WMMA_F32_16X16X4_F32.
// Block = 128 threads = 4 waves; block tile 16 rows x 64 cols (grid.y = col
// half). Tiles staged by the Tensor Data Mover (wave 0 issues 2 DMAs, waits
// on TENSORcnt, workgroup barrier publishes LDS). X rows padded +4 dwords
// in LDS via TDM padding so the 128-dword row stride doesn't alias banks.
// ---------------------------------------------------------------------------
__global__ __launch_bounds__(128) void gemm_wmma_kernel(
    const float* __restrict__ X, const float* __restrict__ W,
    float* __restrict__ H, int n_rows)
{
    __shared__ float xs[16][D_FEAT + 4];
    __shared__ float wsm[D_FEAT][64];

    const int tid = threadIdx.x;
    const int m0  = blockIdx.x * 16;
    const int nb  = blockIdx.y * 64;

    if (tid < 32) {   // wave 0 drives the TDM
        // W tile: rows 0..127, cols nb..nb+63 -> wsm (contiguous rows of 64)
        tdm_load_2d((unsigned)(uintptr_t)&wsm[0][0], W + nb,
                    /*td0=*/(unsigned)(D_FEAT - nb), /*td1=*/(unsigned)D_FEAT,
                    /*stride0=*/(unsigned)D_FEAT,
                    /*tile0=*/64u, /*tile1=*/(unsigned)D_FEAT,
                    0u, 0u, false);
        // X tile: 16 rows starting at m0, all 128 cols; pad 4 dwords after
        // each 128-dword row (interval code 6 = 128 dwords, amount code 3 = 4).
        // Rows past n_rows read as zero via tensor_dim1 OOB semantics.
        tdm_load_2d((unsigned)(uintptr_t)&xs[0][0], X + (size_t)m0 * D_FEAT,
                    /*td0=*/(unsigned)D_FEAT, /*td1=*/(unsigned)(n_rows - m0),
                    /*stride0=*/(unsigned)D_FEAT,
                    /*tile0=*/(unsigned)D_FEAT, /*tile1=*/16u,
                    6u, 3u, true);
        __builtin_amdgcn_s_wait_tensorcnt(0);
    }
    __syncthreads();

    const int wave = tid >> 5;
    const int lane = tid & 31;
    const int half = lane >> 4;           // 0: lanes 0-15, 1: lanes 16-31
    const int l16  = lane & 15;
    const int n0   = wave * 16;           // column offset inside wsm

    v8f c = {};
    #pragma unroll 4
    for (int k = 0; k < D_FEAT; k += 4) {
        // A 16x4 fp32 fragment: VGPR v, lane -> K = k + v + 2*half, M = l16
        v2f a, b;
        a.x = xs[l16][k + 2 * half];
        a.y = xs[l16][k + 2 * half + 1];
        // B 4x16 fp32 fragment: VGPR v, lane -> K = k + v + 2*half, N = l16
        b.x = wsm[k + 2 * half][n0 + l16];
        b.y = wsm[k + 2 * half + 1][n0 + l16];
        c = __builtin_amdgcn_wmma_f32_16x16x4_f32(
                /*neg_a=*/false, a, /*neg_b=*/false, b,
                /*c_mod=*/(short)0, c, /*reuse_a=*/false, /*reuse_b=*/false);
    }

    // D layout: VGPR r, lane -> M = r + 8*half, N = l16
    float* outp = H + (size_t)(m0 + 8 * half) * D_FEAT + nb + n0 + l16;
    if (m0 + 16 <= n_rows) {              // uniform fast path: no per-row guards
        #pragma unroll
        for (int r = 0; r < 8; ++r) outp[(size_t)r * D_FEAT] = c[r];
    } else {
        #pragma unroll
        for (int r = 0; r < 8; ++r)
            if (m0 + 8 * half + r < n_rows) outp[(size_t)r * D_FEAT] = c[r];
    }
}

// ---------------------------------------------------------------------------
// Edge scatter: one wave per edge; each lane handles 4 consecutive features.
// agg[dst] += h[src] * dinv[src]*dinv[dst]  (HW float atomics, L2-resident).
// Edge index is wave-uniform -> readfirstlane forces scalar loads.
// ---------------------------------------------------------------------------
__global__ __launch_bounds__(256) void scatter_edges_kernel(
    const int* __restrict__ src, const int* __restrict__ dst,
    const float* __restrict__ dinv, const float* __restrict__ h,
    float* __restrict__ agg, int n_edges)
{
    int e    = __builtin_amdgcn_readfirstlane((int)((blockIdx.x * blockDim.x + threadIdx.x) >> 5));
    int lane = threadIdx.x & 31;
    if (e >= n_edges) return;
    int s = src[e], d = dst[e];
    float norm = dinv[s] * dinv[d];
    const float4 hv = *(const float4*)(h + (size_t)s * D_FEAT + lane * 4);
    float* a = agg + (size_t)d * D_FEAT + lane * 4;
    unsafeAtomicAdd(a + 0, hv.x * norm);
    unsafeAtomicAdd(a + 1, hv.y * norm);
    unsafeAtomicAdd(a + 2, hv.z * norm);
    unsafeAtomicAdd(a + 3, hv.w * norm);
}

// ---------------------------------------------------------------------------
// Combine: agg = relu(agg + h * dinv^2 + b)   (in place on agg)
// ---------------------------------------------------------------------------
__global__ void combine_relu_kernel(
    const float* __restrict__ h, const float* __restrict__ dinv,
    const float* __restrict__ b, float* __restrict__ agg, int n_nodes)
{
    int i = blockIdx.x * blockDim.x + threadIdx.x;   // float4 index
    if (i >= n_nodes * (D_FEAT / 4)) return;
    int node = i >> 5;
    int c4   = (i & 31) * 4;
    float di = dinv[node];
    float d2 = di * di;
    float4 hv = *(const float4*)(h + (size_t)node * D_FEAT + c4);
    float4 av = *(const float4*)(agg + (size_t)node * D_FEAT + c4);
    float4 bv = *(const float4*)(b + c4);
    float4 r;
    r.x = fmaxf(fmaf(hv.x, d2, av.x) + bv.x, 0.f);
    r.y = fmaxf(fmaf(hv.y, d2, av.y) + bv.y, 0.f);
    r.z = fmaxf(fmaf(hv.z, d2, av.z) + bv.z, 0.f);
    r.w = fmaxf(fmaf(hv.w, d2, av.w) + bv.w, 0.f);
    *(float4*)(agg + (size_t)node * D_FEAT + c4) = r;
}

// ---------------------------------------------------------------------------
// Readout: out[i] = dot(h[i,:], Wr) + br   — one wave per node, shuffle reduce.
// ---------------------------------------------------------------------------
__global__ __launch_bounds__(256) void readout_kernel(
    const float* __restrict__ h, const float* __restrict__ Wr,
    const float* __restrict__ br, float* __restrict__ out, int n_nodes)
{
    int node = (blockIdx.x * blockDim.x + threadIdx.x) >> 5;
    int lane = threadIdx.x & 31;
    if (node >= n_nodes) return;
    float4 hv = *(const float4*)(h + (size_t)node * D_FEAT + lane * 4);
    float4 wv = *(const float4*)(Wr + lane * 4);
    float p = hv.x * wv.x + hv.y * wv.y + hv.z * wv.z + hv.w * wv.w;
    #pragma unroll
    for (int off = 16; off > 0; off >>= 1) p += __shfl_xor(p, off, 32);
    if (lane == 0) out[node] = p + br[0];
}

// ---------------------------------------------------------------------------
extern "C" void kernel_launch(void* const* d_in, const int* in_sizes, int n_in,
                              void* d_out, int out_size, void* d_ws, size_t ws_size,
                              hipStream_t stream) {
    const float* x  = (const float*)d_in[0];
    const int*   ei = (const int*)d_in[1];
    const float* W1 = (const float*)d_in[2];
    const float* b1 = (const float*)d_in[3];
    const float* W2 = (const float*)d_in[4];
    const float* b2 = (const float*)d_in[5];
    const float* Wr = (const float*)d_in[6];
    const float* br = (const float*)d_in[7];
    float* out = (float*)d_out;

    const int N = in_sizes[0] / D_FEAT;
    const int E = in_sizes[1] / 2;
    const int* srcI = ei;
    const int* dstI = ei + E;

    // Workspace layout: dinv[N] | hA[N*128] | hB[N*128]
    char* ws = (char*)d_ws;
    size_t off = 0;
    float* dinv = (float*)(ws + off);
    off += ((size_t)N * sizeof(float) + 255) & ~(size_t)255;
    float* hA = (float*)(ws + off);
    off += ((size_t)N * D_FEAT * sizeof(float) + 255) & ~(size_t)255;
    float* hB = (float*)(ws + off);

    const int nodeBlocks  = (N + 255) / 256;
    const int edgeBlocks  = (E + 255) / 256;
    const int feat4Blocks = (N * (D_FEAT / 4) + 255) / 256;
    const int waveBlocksE = (int)(((size_t)E * 32 + 255) / 256);
    const int waveBlocksN = (int)(((size_t)N * 32 + 255) / 256);
    const dim3 gemmGrid((N + 15) / 16, 2);

    // --- degree / normalization (shared by both layers) ---
    init_ones_kernel<<<nodeBlocks, 256, 0, stream>>>(dinv, N);
    count_deg_kernel<<<edgeBlocks, 256, 0, stream>>>(dstI, dinv, E);
    rsqrt_kernel<<<nodeBlocks, 256, 0, stream>>>(dinv, N);

    // --- layer 1 ---
    gemm_wmma_kernel<<<gemmGrid, 128, 0, stream>>>(x, W1, hA, N);
    zero4_kernel<<<feat4Blocks, 256, 0, stream>>>((float4*)hB, N * (D_FEAT / 4));
    scatter_edges_kernel<<<waveBlocksE, 256, 0, stream>>>(srcI, dstI, dinv, hA, hB, E);
    combine_relu_kernel<<<feat4Blocks, 256, 0, stream>>>(hA, dinv, b1, hB, N);

    // --- layer 2 ---
    gemm_wmma_kernel<<<gemmGrid, 128, 0, stream>>>(hB, W2, hA, N);
    zero4_kernel<<<feat4Blocks, 256, 0, stream>>>((float4*)hB, N * (D_FEAT / 4));
    scatter_edges_kernel<<<waveBlocksE, 256, 0, stream>>>(srcI, dstI, dinv, hA, hB, E);
    combine_relu_kernel<<<feat4Blocks, 256, 0, stream>>>(hA, dinv, b2, hB, N);

    // --- readout ---
    readout_kernel<<<waveBlocksN, 256, 0, stream>>>(hB, Wr, br, out, N);
}